// YOLOv1Loss_27917287424001
// MI455X (gfx1250) — compile-verified
//
#include <hip/hip_runtime.h>
#include <hip/hip_bf16.h>

// ---------------------------------------------------------------------------
// YOLOv1 loss, MI455X (gfx1250, wave32).
// Bandwidth-bound streaming reduction (roofline ~8.3 us @ 23.3 TB/s):
//   - async global->LDS staging (GLOBAL_LOAD_ASYNC_TO_LDS_B128, ASYNCcnt):
//     data flows HBM/L2 -> LDS with no VGPR round-trip
//   - branch-free per-cell loss (EXEC stays all-ones)
//   - exact f32 wave reduction via V_WMMA_F32_16X16X4_F32
// ---------------------------------------------------------------------------

typedef __attribute__((ext_vector_type(2))) float v2f;
typedef __attribute__((ext_vector_type(8))) float v8f;

#define TILE_CELLS   256                    // cells per block (= blockDim.x)
#define TILE_FLOATS  (TILE_CELLS * 30)      // 7680 floats per tensor per tile
#define TILE_CHUNKS  (TILE_FLOATS / 4)      // 1920 float4 chunks per tensor

#if defined(__has_builtin)
#if __has_builtin(__builtin_amdgcn_global_load_async_to_lds_b128)
#define USE_ASYNC_LDS 1
#endif
#endif

#if defined(USE_ASYNC_LDS)
// Param 0 (per hipcc diagnostic): non-const pointer to gcc-vector int4 in the
// global address space; mirror the type for the LDS-side argument in AS(3).
typedef int v4i_vs __attribute__((vector_size(4 * sizeof(int))));
typedef __attribute__((address_space(1))) v4i_vs gv4i_t;
typedef __attribute__((address_space(3))) v4i_vs lv4i_t;

__device__ __forceinline__ void async_copy16(const void* gsrc, void* ldst) {
    // (global src, LDS dst, imm offset, imm cpol=RT)
    __builtin_amdgcn_global_load_async_to_lds_b128(
        (gv4i_t*)gsrc, (lv4i_t*)ldst, 0, 0);
}

__device__ __forceinline__ void async_drain() {
#if __has_builtin(__builtin_amdgcn_s_wait_asynccnt)
    __builtin_amdgcn_s_wait_asynccnt(0);
#else
    asm volatile("s_wait_asynccnt 0x0" ::: "memory");
#endif
}
#endif

// Exact f32 sum of all 32 lanes' x, broadcast to every lane.
// A = 16x4 f32 with per-lane {x, 0}: A[m][0]=x(lane m), A[m][2]=x(lane m+16).
// B = ones(4x16)  =>  D[m][n] = x_m + x_{m+16} for every n.
// Sum of the 8 accumulator VGPRs + shfl_xor(16) -> full wave sum, exact f32.
__device__ __forceinline__ float wave_sum_wmma(float x) {
    v2f a; a[0] = x;    a[1] = 0.0f;
    v2f b; b[0] = 1.0f; b[1] = 1.0f;
    v8f c = {};
    c = __builtin_amdgcn_wmma_f32_16x16x4_f32(
            /*neg_a=*/false, a, /*neg_b=*/false, b,
            /*c_mod=*/(short)0, c, /*reuse_a=*/false, /*reuse_b=*/false);
    float s = ((c[0] + c[1]) + (c[2] + c[3])) + ((c[4] + c[5]) + (c[6] + c[7]));
    s += __shfl_xor(s, 16, 32);
    return s;
}

__device__ __forceinline__ float iou_f(float ax, float ay, float aw, float ah,
                                       float bx, float by, float bw, float bh) {
    float ax1 = ax - aw * 0.5f, ay1 = ay - ah * 0.5f;
    float ax2 = ax + aw * 0.5f, ay2 = ay + ah * 0.5f;
    float bx1 = bx - bw * 0.5f, by1 = by - bh * 0.5f;
    float bx2 = bx + bw * 0.5f, by2 = by + bh * 0.5f;
    float iw = fmaxf(fminf(ax2, bx2) - fmaxf(ax1, bx1), 0.0f);
    float ih = fmaxf(fminf(ay2, by2) - fmaxf(ay1, by1), 0.0f);
    float inter = iw * ih;
    float uni   = aw * ah + bw * bh - inter;
    return inter / (uni + 1e-10f);
}

__global__ __launch_bounds__(TILE_CELLS)
void yolo_partial_kernel(const float* __restrict__ preds,
                         const float* __restrict__ labels,
                         float* __restrict__ partials,
                         int cells) {
    __shared__ float sp[TILE_FLOATS];   // 30 KB preds tile
    __shared__ float sl[TILE_FLOATS];   // 30 KB labels tile
    __shared__ float sred[8];           // per-wave partial sums

    const int  tid  = threadIdx.x;
    const long base = (long)blockIdx.x * TILE_CHUNKS;
    const long maxChunk = ((long)cells * 30) / 4 - 1;   // 30*cells divisible by 4 here

    const float4* gp = reinterpret_cast<const float4*>(preds);
    const float4* gl = reinterpret_cast<const float4*>(labels);
    float4* s4p = reinterpret_cast<float4*>(sp);
    float4* s4l = reinterpret_cast<float4*>(sl);

    // Coalesced staging: contiguous float4 per lane (512 B / wave / instruction).
    // Iteration count is uniform within each wave -> EXEC all-ones per issue.
    for (int i = tid; i < TILE_CHUNKS; i += TILE_CELLS) {
        long g = base + i;
        if (g > maxChunk) g = maxChunk;   // tail-safe clamp (no divergence)
#if defined(USE_ASYNC_LDS)
        async_copy16(gp + g, s4p + i);    // HBM/L2 -> LDS, no VGPR round-trip
        async_copy16(gl + g, s4l + i);
#else
        s4p[i] = gp[g];
        s4l[i] = gl[g];
#endif
    }
#if defined(USE_ASYNC_LDS)
    async_drain();                        // this wave's ASYNCcnt == 0
#endif
    __syncthreads();                      // all waves' tiles visible

    // --- per-cell loss (branch-free; EXEC stays all-ones for WMMA) ---------
    const int   cell = blockIdx.x * TILE_CELLS + tid;
    const float w    = (cell < cells) ? 1.0f : 0.0f;
    const float* P = sp + tid * 30;     // lane stride 30 -> 32 distinct banks
    const float* L = sl + tid * 30;

    const float obj = (L[4] == 1.0f) ? 1.0f : 0.0f;

    const float iou1 = iou_f(P[0], P[1], P[2], P[3], L[0], L[1], L[2], L[3]);
    const float iou2 = iou_f(P[5], P[6], P[7], P[8], L[0], L[1], L[2], L[3]);
    const bool  b1   = iou1 > iou2;

    const float px = b1 ? P[0] : P[5],  py = b1 ? P[1] : P[6];
    const float pw = b1 ? P[2] : P[7],  ph = b1 ? P[3] : P[8];
    const float lx = b1 ? L[0] : L[5],  ly = b1 ? L[1] : L[6];
    const float lw = b1 ? L[2] : L[7],  lh = b1 ? L[3] : L[8];
    const float conf_r = b1 ? P[4] : P[9];
    const float conf_o = b1 ? P[9] : P[4];
    const float iou_r  = b1 ? iou1 : iou2;
    const float iou_o  = b1 ? iou2 : iou1;

    const float dx = px - lx, dy = py - ly;
    float loss = 5.0f * obj * (dx * dx + dy * dy);             // loss_xy

    const float sw = sqrtf(pw) - sqrtf(lw);
    const float sh = sqrtf(ph) - sqrtf(lh);
    loss += obj * (sw * sw + sh * sh);                         // loss_wh

    const float dcr = conf_r - iou_r;
    loss += obj * dcr * dcr;                                   // loss_obj

    const float dco = conf_o - iou_o;
    loss += 0.5f * obj * dco * dco;                            // loss_noobj_in

    loss += 0.5f * (1.0f - obj) * (P[4] * P[4] + P[9] * P[9]); // loss_noobj_out

    float cl = 0.0f;
    #pragma unroll
    for (int i = 10; i < 30; ++i) {
        const float d = P[i] - L[i];
        cl += d * d;
    }
    loss += obj * cl;                                          // loss_cls

    loss *= w;

    // --- hierarchical reduction: WMMA wave sum -> LDS -> block partial -----
    const float wsum = wave_sum_wmma(loss);
    const int wid = tid >> 5;
    if ((tid & 31) == 0) sred[wid] = wsum;
    __syncthreads();
    if (tid == 0) {
        float t = 0.0f;
        #pragma unroll
        for (int v = 0; v < 8; ++v) t += sred[v];
        partials[blockIdx.x] = t;
    }
}

__global__ __launch_bounds__(256)
void yolo_finalize_kernel(const float* __restrict__ partials, int n,
                          float* __restrict__ out, float inv_batch) {
    __shared__ float sred[8];
    float acc = 0.0f;
    for (int i = threadIdx.x; i < n; i += 256) acc += partials[i];
    const float wsum = wave_sum_wmma(acc);
    const int wid = threadIdx.x >> 5;
    if ((threadIdx.x & 31) == 0) sred[wid] = wsum;
    __syncthreads();
    if (threadIdx.x == 0) {
        float t = 0.0f;
        #pragma unroll
        for (int v = 0; v < 8; ++v) t += sred[v];
        out[0] = t * inv_batch;
    }
}

extern "C" void kernel_launch(void* const* d_in, const int* in_sizes, int n_in,
                              void* d_out, int out_size, void* d_ws, size_t ws_size,
                              hipStream_t stream) {
    const float* preds  = (const float*)d_in[0];
    const float* labels = (const float*)d_in[1];
    float* out  = (float*)d_out;
    float* part = (float*)d_ws;

    const long total_floats = (long)in_sizes[0];       // BATCH*7*7*30
    const int  cells  = (int)(total_floats / 30);      // 802,816
    const int  batch  = (int)(total_floats / 1470);    // 16,384
    const int  blocks = (cells + TILE_CELLS - 1) / TILE_CELLS;  // 3,136

    yolo_partial_kernel<<<blocks, TILE_CELLS, 0, stream>>>(preds, labels, part, cells);
    yolo_finalize_kernel<<<1, 256, 0, stream>>>(part, blocks, out,
                                                1.0f / (float)batch);
}